// TripletRankingLoss_42365557408211
// MI455X (gfx1250) — compile-verified
//
#include <hip/hip_runtime.h>

// TripletRankingLoss on MI455X (gfx1250, wave32).
//   pred: [B=16384, D=256] f32, idx arrays: [T=262144] int
// Strategy:
//   K1: invn[b] = 1/max(||pred[b]||, 1e-6)        (one wave per row)
//   K2: one wave per 16-triplet tile; dot products via V_WMMA_F32_16X16X4_F32,
//       diagonal of the 16x16 product = the 16 needed dots. Per-block partial sum.
//   K3: deterministic single-block tree reduction -> mean.

typedef float v2f __attribute__((ext_vector_type(2)));
typedef float v8f __attribute__((ext_vector_type(8)));

#define D_DIM   256
#define MARGIN  0.1f
#define EPS_F   1e-6f

__global__ void invnorm_kernel(const float* __restrict__ pred,
                               float* __restrict__ invn, int B) {
    int wave = (blockIdx.x * blockDim.x + threadIdx.x) >> 5;
    int lane = threadIdx.x & 31;
    if (wave >= B) return;
    const float* row = pred + (size_t)wave * D_DIM + lane * 8;
    float4 x0 = *(const float4*)(row);
    float4 x1 = *(const float4*)(row + 4);
    float s = x0.x * x0.x + x0.y * x0.y + x0.z * x0.z + x0.w * x0.w
            + x1.x * x1.x + x1.y * x1.y + x1.z * x1.z + x1.w * x1.w;
    #pragma unroll
    for (int m = 16; m >= 1; m >>= 1) s += __shfl_xor(s, m, 32);
    if (lane == 0) invn[wave] = 1.0f / fmaxf(sqrtf(s), EPS_F);
}

__global__ void __launch_bounds__(256)
triplet_wmma_kernel(const float* __restrict__ pred,
                    const int* __restrict__ aidx,
                    const int* __restrict__ pidx,
                    const int* __restrict__ nidx,
                    const float* __restrict__ invn,
                    float* __restrict__ partials) {
    const int lane        = threadIdx.x & 31;
    const int waveInBlock = threadIdx.x >> 5;
    const int tile        = blockIdx.x * (blockDim.x >> 5) + waveInBlock;
    const int t0          = tile * 16;

    // Fragment loader lanes: lane L supplies row (L&15), d-offset (L>>4)*2.
    const int iload = lane & 15;
    const int t     = t0 + iload;
    const int ar = aidx[t], pr = pidx[t], nr = nidx[t];
    const int dof = (lane >> 4) << 1;
    const float* pa = pred + (size_t)ar * D_DIM + dof;
    const float* pp = pred + (size_t)pr * D_DIM + dof;
    const float* pn = pred + (size_t)nr * D_DIM + dof;

    v8f accP = {};
    v8f accN = {};
    // D = 256 -> 64 K-chunks of 4. A = anchors(16xK), B = pos/neg (Kx16).
    #pragma unroll 4
    for (int k = 0; k < D_DIM / 4; ++k) {
        v2f va = *(const v2f*)(pa + k * 4);
        v2f vb = *(const v2f*)(pp + k * 4);
        v2f vc = *(const v2f*)(pn + k * 4);
        accP = __builtin_amdgcn_wmma_f32_16x16x4_f32(
            false, va, false, vb, (short)0, accP, false, false);
        accN = __builtin_amdgcn_wmma_f32_16x16x4_f32(
            false, va, false, vc, (short)0, accN, false, false);
    }

    // Diagonal (i,i): i<8 -> VGPR i @ lane i; i>=8 -> VGPR i-8 @ lane i+16.
    float dP = 0.f, dN = 0.f;
    const int r = lane & 7;
    #pragma unroll
    for (int q = 0; q < 8; ++q) {
        if (r == q) { dP = accP[q]; dN = accN[q]; }
    }

    float val = 0.f;
    if ((lane & 8) == 0) {                     // owner lanes: 0-7, 16-23
        const int i  = (lane & 7) + ((lane >> 4) << 3);
        const int tt = t0 + i;
        const int a2 = aidx[tt], p2 = pidx[tt], n2 = nidx[tt];
        const float ia = invn[a2];
        const float c  = dP * ia * invn[p2] - dN * ia * invn[n2] + MARGIN;
        val = fmaxf(c, 0.f);
    }

    // Wave reduce, then fixed-order block reduce (deterministic, no atomics).
    #pragma unroll
    for (int m = 16; m >= 1; m >>= 1) val += __shfl_xor(val, m, 32);
    __shared__ float sm[8];
    if (lane == 0) sm[waveInBlock] = val;
    __syncthreads();
    if (threadIdx.x == 0) {
        float s = 0.f;
        #pragma unroll
        for (int w = 0; w < 8; ++w) s += sm[w];
        partials[blockIdx.x] = s;
    }
}

__global__ void reduce_kernel(const float* __restrict__ partials, int n,
                              float* __restrict__ out, float invT) {
    __shared__ float sm[256];
    float s = 0.f;
    for (int i = threadIdx.x; i < n; i += 256) s += partials[i];
    sm[threadIdx.x] = s;
    __syncthreads();
    #pragma unroll
    for (int step = 128; step >= 1; step >>= 1) {
        if (threadIdx.x < step) sm[threadIdx.x] += sm[threadIdx.x + step];
        __syncthreads();
    }
    if (threadIdx.x == 0) out[0] = sm[0] * invT;
}

extern "C" void kernel_launch(void* const* d_in, const int* in_sizes, int n_in,
                              void* d_out, int out_size, void* d_ws, size_t ws_size,
                              hipStream_t stream) {
    const float* pred = (const float*)d_in[0];
    const int*   aidx = (const int*)d_in[1];
    const int*   pidx = (const int*)d_in[2];
    const int*   nidx = (const int*)d_in[3];

    const int B = in_sizes[0] / D_DIM;     // 16384
    const int T = in_sizes[1];             // 262144

    float* invn     = (float*)d_ws;        // B floats
    float* partials = invn + B;            // tiles/8 floats

    const int blocks1 = (B + 7) / 8;       // 8 rows (waves) per 256-thread block
    invnorm_kernel<<<blocks1, 256, 0, stream>>>(pred, invn, B);

    const int tiles   = T / 16;            // 16 triplets per wave
    const int blocks2 = tiles / 8;         // 8 waves per block
    triplet_wmma_kernel<<<blocks2, 256, 0, stream>>>(pred, aidx, pidx, nidx,
                                                     invn, partials);

    reduce_kernel<<<1, 256, 0, stream>>>(partials, blocks2, (float*)d_out,
                                         1.0f / (float)T);
}